// LanguageModel_249108103775
// MI455X (gfx1250) — compile-verified
//
#include <hip/hip_runtime.h>
#include <hip/hip_bf16.h>
#include <math.h>
#include <stdint.h>

typedef float v2f __attribute__((ext_vector_type(2)));
typedef float v8f __attribute__((ext_vector_type(8)));
typedef unsigned int u32x4 __attribute__((ext_vector_type(4)));
typedef int i32x4 __attribute__((ext_vector_type(4)));
typedef int i32x8 __attribute__((ext_vector_type(8)));

constexpr int T  = 128;
constexpr int B  = 32;
constexpr int V  = 32000;
constexpr int E  = 1024;
constexpr int H  = 1024;
constexpr int FH = 4 * H;   // 4096
constexpr int TB = T * B;   // 4096
constexpr int KC = 32;      // K-chunk staged in LDS by the TDM (32x64 f32 = 8 KB)

__device__ __forceinline__ float sigmoidf_(float x) {
    return 1.0f / (1.0f + __expf(-x));
}

__device__ __forceinline__ v8f wmma_f32(v2f a, v2f b, v8f c) {
    return __builtin_amdgcn_wmma_f32_16x16x4_f32(
        false, a, false, b, (short)0, c, false, false);
}

// Issue one TDM descriptor: copy a [KC x 64] f32 tile of W_h2o (row stride V)
// into LDS. Tracked by TENSORcnt. (ISA ch.8 D# layout: count=1, type=2,
// data_size=4B, tile_dim0=64, tile_dim1=KC, tensor_dim0_stride=V.)
__device__ __forceinline__ void tdm_load_chunk(const float* gsrc, float* ldsdst) {
    const unsigned long long ga = (unsigned long long)(uintptr_t)gsrc;
    u32x4 g0;
    g0.x = 1u;                                            // count=1, user mode
    g0.y = (unsigned)(uintptr_t)ldsdst;                   // LDS byte address
    g0.z = (unsigned)(ga & 0xFFFFFFFFull);                // global_addr[31:0]
    g0.w = (unsigned)((ga >> 32) & 0x01FFFFFFull) | 0x80000000u; // type=2
    i32x8 g1;
    g1[0] = 0x00020000;                                   // data_size = 4B
    g1[1] = (int)(((unsigned)V & 0xFFFFu) << 16);         // tensor_dim0 lo
    g1[2] = (int)((((unsigned)V >> 16) & 0xFFFFu)         // tensor_dim0 hi
                  | (((unsigned)H & 0xFFFFu) << 16));     // tensor_dim1 lo
    g1[3] = (int)((((unsigned)H >> 16) & 0xFFFFu)         // tensor_dim1 hi
                  | (64u << 16));                         // tile_dim0 = 64
    g1[4] = KC;                                           // tile_dim1; tile_dim2=0
    g1[5] = V;                                            // tensor_dim0_stride lo
    g1[6] = 0;
    g1[7] = 0;
    const i32x4 gz4 = {0, 0, 0, 0};
    const i32x8 gz8 = {0, 0, 0, 0, 0, 0, 0, 0};
    __builtin_amdgcn_tensor_load_to_lds(g0, g1, gz4, gz4, gz8, 0);
}

// ---------------------------------------------------------------------------
// Zero-fill (c0 and h0 state buffers) — must run every kernel_launch call.
// ---------------------------------------------------------------------------
__global__ void zero_kernel(float* __restrict__ p, int n) {
    int i = blockIdx.x * blockDim.x + threadIdx.x;
    if (i < n) p[i] = 0.0f;
}

// ---------------------------------------------------------------------------
// Phase 1: preact[r, j] = emb[tokens[r]] @ Wx + b   for all r in [0, T*B)
// One wave per (16-row M tile) x (64-col N group); 4 f32 WMMA accumulators.
// Output stream (64 MB) is stored non-temporally so it does not evict the
// L2-resident weights.
// ---------------------------------------------------------------------------
__global__ void preact_kernel(const int*   __restrict__ tokens,
                              const float* __restrict__ emb,
                              const float* __restrict__ Wx,
                              const float* __restrict__ bias,
                              float*       __restrict__ preact) {
    const int wave = (blockIdx.x * blockDim.x + threadIdx.x) >> 5;
    const int lane = threadIdx.x & 31;
    const int half = lane >> 4;   // 0: lanes 0-15, 1: lanes 16-31
    const int lm   = lane & 15;

    const int NG    = FH / 64;    // 64 column groups
    const int mtile = wave / NG;
    const int ng    = wave % NG;
    const int m0 = mtile * 16;
    const int n0 = ng * 64;

    // Gathered A row: lane holds row m0+lm of X = emb[tokens]
    const float* arow = emb + (long)tokens[m0 + lm] * E;

    v8f acc[4];
    #pragma unroll
    for (int nt = 0; nt < 4; ++nt) {
        const float bv = bias[n0 + nt * 16 + lm];   // N fixed per lane
        acc[nt] = (v8f){bv, bv, bv, bv, bv, bv, bv, bv};
    }

    for (int k = 0; k < E; k += 4) {
        const v2f a = *(const v2f*)(arow + k + 2 * half);
        #pragma unroll
        for (int nt = 0; nt < 4; ++nt) {
            const int col = n0 + nt * 16 + lm;
            v2f bv;
            bv.x = Wx[(long)(k + 2 * half + 0) * FH + col];
            bv.y = Wx[(long)(k + 2 * half + 1) * FH + col];
            acc[nt] = wmma_f32(a, bv, acc[nt]);
        }
    }

    #pragma unroll
    for (int nt = 0; nt < 4; ++nt)
        #pragma unroll
        for (int r = 0; r < 8; ++r)
            __builtin_nontemporal_store(
                acc[nt][r],
                &preact[(long)(m0 + r + 8 * half) * FH + n0 + nt * 16 + lm]);
}

// ---------------------------------------------------------------------------
// Phase 2: one recurrent step (gates = preact_t + h_prev @ Wh; LSTM update).
// Each wave owns a (16 batch) x (16 h-col) tile and computes ALL 4 gate tiles
// so the nonlinearity + state update fuse locally. 128 waves per step.
// Wh (16 MB) stays L2-resident across all 128 sequential launches.
// ---------------------------------------------------------------------------
__global__ void lstm_step_kernel(const float* __restrict__ preact_t,  // [B, 4H]
                                 const float* __restrict__ Wh,        // [H, 4H]
                                 const float* __restrict__ hprev,     // [B, H]
                                 float*       __restrict__ hout,      // [B, H]
                                 float*       __restrict__ cbuf) {    // [B, H]
    const int wave = (blockIdx.x * blockDim.x + threadIdx.x) >> 5;
    const int lane = threadIdx.x & 31;
    const int half = lane >> 4;
    const int lm   = lane & 15;

    const int NT    = H / 16;     // 64 h-column tiles
    const int mtile = wave / NT;  // 0..1
    const int ht    = wave % NT;
    const int m0 = mtile * 16;
    const int h0 = ht * 16;

    // Accumulators = the 4 gate tiles (i, f, g, o), seeded from preact_t.
    v8f acc[4];
    #pragma unroll
    for (int g = 0; g < 4; ++g)
        #pragma unroll
        for (int r = 0; r < 8; ++r)
            acc[g][r] = preact_t[(long)(m0 + r + 8 * half) * FH + g * H + h0 + lm];

    const float* arow = hprev + (long)(m0 + lm) * H;
    for (int k = 0; k < H; k += 4) {
        const v2f a = *(const v2f*)(arow + k + 2 * half);
        #pragma unroll
        for (int g = 0; g < 4; ++g) {
            const int col = g * H + h0 + lm;
            v2f bv;
            bv.x = Wh[(long)(k + 2 * half + 0) * FH + col];
            bv.y = Wh[(long)(k + 2 * half + 1) * FH + col];
            acc[g] = wmma_f32(a, bv, acc[g]);
        }
    }

    // Fused pointwise LSTM update on this (16b x 16h) tile.
    #pragma unroll
    for (int r = 0; r < 8; ++r) {
        const int  bm  = m0 + r + 8 * half;
        const long idx = (long)bm * H + h0 + lm;
        const float iv = sigmoidf_(acc[0][r]);
        const float fv = sigmoidf_(acc[1][r]);
        const float gv = tanhf(acc[2][r]);
        const float ov = sigmoidf_(acc[3][r]);
        const float c  = fv * cbuf[idx] + iv * gv;
        cbuf[idx] = c;
        hout[idx] = ov * tanhf(c);
    }
}

// ---------------------------------------------------------------------------
// Phase 3: out = hstates @ W_h2o + b_o   ([4096,1024] x [1024,32000]).
// Block = 4 waves covering 128M x 64N; each wave computes 32M x 64N
// (8 accumulators), so each B fragment feeds two WMMAs. The shared B chunk
// (KC x 64 f32 = 8 KB) is staged into LDS by the Tensor Data Mover with
// DOUBLE BUFFERING: the DMA for chunk i+1 overlaps the WMMA work on chunk i
// (exactly one descriptor in flight -> s_wait_tensorcnt(0) at loop top).
// The 512 MB output is stored non-temporally to keep W_h2o L2-resident.
// ---------------------------------------------------------------------------
__global__ void __launch_bounds__(128)
proj_kernel(const float* __restrict__ hstates,  // [TB, H]
            const float* __restrict__ W_h2o,    // [H, V]
            const float* __restrict__ b_o,      // [V]
            float*       __restrict__ out) {    // [TB, V]
    __shared__ float ldsB[2][KC * 64];          // 16 KB double buffer

    const int tid  = threadIdx.x;
    const int wid  = tid >> 5;
    const int lane = tid & 31;
    const int half = lane >> 4;
    const int lm   = lane & 15;

    const int n0    = blockIdx.x * 64;             // 500 column groups
    const int mbase = blockIdx.y * 128 + wid * 32; // block M=128, wave M=32

    v8f acc[2][4];
    #pragma unroll
    for (int nt = 0; nt < 4; ++nt) {
        const float bv = b_o[n0 + nt * 16 + lm];
        acc[0][nt] = (v8f){bv, bv, bv, bv, bv, bv, bv, bv};
        acc[1][nt] = acc[0][nt];
    }

    const float* arow0 = hstates + (long)(mbase + lm) * H;
    const float* arow1 = hstates + (long)(mbase + 16 + lm) * H;
    const float* wcol  = W_h2o + n0;

    constexpr int NCHUNK = H / KC;  // 32

    // Prologue: start DMA of chunk 0 into buffer 0.
    if (wid == 0) tdm_load_chunk(wcol, ldsB[0]);

    for (int i = 0; i < NCHUNK; ++i) {
        if (wid == 0) __builtin_amdgcn_s_wait_tensorcnt(0);  // chunk i landed
        __syncthreads();  // chunk i visible; readers of buf[(i+1)&1] (chunk i-1) done
        if (wid == 0 && i + 1 < NCHUNK)
            tdm_load_chunk(wcol + (long)(i + 1) * KC * V, ldsB[(i + 1) & 1]);

        const float* Bb = ldsB[i & 1];
        const int k0 = i * KC;
        #pragma unroll
        for (int kk = 0; kk < KC; kk += 4) {
            const int k = k0 + kk;
            const v2f a0 = *(const v2f*)(arow0 + k + 2 * half);
            const v2f a1 = *(const v2f*)(arow1 + k + 2 * half);
            #pragma unroll
            for (int nt = 0; nt < 4; ++nt) {
                const int cc = nt * 16 + lm;
                v2f bv;
                bv.x = Bb[(kk + 2 * half + 0) * 64 + cc];
                bv.y = Bb[(kk + 2 * half + 1) * 64 + cc];
                acc[0][nt] = wmma_f32(a0, bv, acc[0][nt]);
                acc[1][nt] = wmma_f32(a1, bv, acc[1][nt]);
            }
        }
    }

    #pragma unroll
    for (int mt = 0; mt < 2; ++mt)
        #pragma unroll
        for (int nt = 0; nt < 4; ++nt)
            #pragma unroll
            for (int r = 0; r < 8; ++r)
                __builtin_nontemporal_store(
                    acc[mt][nt][r],
                    &out[(long)(mbase + mt * 16 + r + 8 * half) * V
                         + n0 + nt * 16 + lm]);
}

// ---------------------------------------------------------------------------
// Host launcher. Inputs (reference order):
//   0 tokens [T,B] i32, 1 emb [V,E] f32, 2 Wx [E,4H], 3 Wh [H,4H], 4 b [4H],
//   5 W_h2o [H,V], 6 b_o [V].  Output: [T,B,V] f32.
// Workspace: preact (64 MB) | hstates (16 MB) | c (128 KB) | h0 zeros (128 KB)
// ---------------------------------------------------------------------------
extern "C" void kernel_launch(void* const* d_in, const int* in_sizes, int n_in,
                              void* d_out, int out_size, void* d_ws, size_t ws_size,
                              hipStream_t stream) {
    const int*   tokens = (const int*)  d_in[0];
    const float* emb    = (const float*)d_in[1];
    const float* Wx     = (const float*)d_in[2];
    const float* Wh     = (const float*)d_in[3];
    const float* b      = (const float*)d_in[4];
    const float* W_h2o  = (const float*)d_in[5];
    const float* b_o    = (const float*)d_in[6];
    float* out = (float*)d_out;

    float* preact  = (float*)d_ws;                    // TB * FH
    float* hstates = preact  + (size_t)TB * FH;       // TB * H
    float* cbuf    = hstates + (size_t)TB * H;        // B * H
    float* h0      = cbuf    + (size_t)B * H;         // B * H (zeros)

    // Zero c0 and h0 (contiguous 2*B*H floats).
    zero_kernel<<<(2 * B * H + 255) / 256, 256, 0, stream>>>(cbuf, 2 * B * H);

    // Phase 1: (TB/16)*(FH/64) = 16384 waves, 4 waves/block.
    preact_kernel<<<(TB / 16) * (FH / 64) / 4, 128, 0, stream>>>(
        tokens, emb, Wx, b, preact);

    // Phase 2: 128 sequential steps; 128 waves each (32 blocks x 4 waves).
    for (int t = 0; t < T; ++t) {
        const float* hprev = (t == 0) ? h0 : (hstates + (size_t)(t - 1) * B * H);
        float*       hout  = hstates + (size_t)t * B * H;
        lstm_step_kernel<<<(2 * (H / 16)) / 4, 128, 0, stream>>>(
            preact + (size_t)t * B * FH, Wh, hprev, hout, cbuf);
    }

    // Phase 3: grid (V/64, TB/128) = (500, 32) blocks of 128 threads.
    dim3 pgrid(V / 64, TB / 128);
    proj_kernel<<<pgrid, 128, 0, stream>>>(hstates, W_h2o, b_o, out);
}